// Attention_48258252537865
// MI455X (gfx1250) — compile-verified
//
#include <hip/hip_runtime.h>
#include <hip/hip_bf16.h>
#include <stdint.h>

typedef float v2f __attribute__((ext_vector_type(2)));
typedef float v8f __attribute__((ext_vector_type(8)));
typedef unsigned int u32x4 __attribute__((ext_vector_type(4)));
typedef int i32x8 __attribute__((ext_vector_type(8)));
typedef int i32x4 __attribute__((ext_vector_type(4)));

#define S_LEN 8192
#define H2 1024
#define A_DIM 2048
#define E_DIM 768
#define M_TILE 32
#define LDS_ROW 1028                      // 1024 dwords + 4 pad dwords (TDM pad: 1 DW per 256 DW)
#define TILE_DW (M_TILE * LDS_ROW)        // 32896 dwords = 131584 B
#define K2_LDS_BYTES (TILE_DW * 4 + M_TILE * 4)

#ifndef ATTN_HAVE_TDM
#if defined(__has_builtin)
#if __has_builtin(__builtin_amdgcn_tensor_load_to_lds)
#define ATTN_HAVE_TDM 1
#endif
#endif
#endif

// Native gfx1250 V_TANH_F32 if the builtin exists; safe fallback otherwise.
#if defined(__has_builtin)
#if __has_builtin(__builtin_amdgcn_tanhf)
#define ATTN_TANH(x) __builtin_amdgcn_tanhf(x)
#endif
#endif
#ifndef ATTN_TANH
#define ATTN_TANH(x) tanhf(x)
#endif

// ---------------------------------------------------------------------------
// K1: cbias[tag][a] = b_sent[a] + b_ctx[a] + sum_e ctx[e] * W_ctx[e][a]
// ---------------------------------------------------------------------------
__global__ void attn_cbias_kernel(const float* __restrict__ ctx_p,
                                  const float* __restrict__ ctx_c,
                                  const float* __restrict__ ctx_h,
                                  const float* __restrict__ Wc_p,
                                  const float* __restrict__ Wc_c,
                                  const float* __restrict__ Wc_h,
                                  const float* __restrict__ bs_p,
                                  const float* __restrict__ bs_c,
                                  const float* __restrict__ bs_h,
                                  const float* __restrict__ bc_p,
                                  const float* __restrict__ bc_c,
                                  const float* __restrict__ bc_h,
                                  float* __restrict__ cbias) {
  const int tag = blockIdx.x >> 3;                   // 8 blocks of 256 cover A=2048
  const int a = ((blockIdx.x & 7) << 8) + threadIdx.x;
  const float* ctx = (tag == 0) ? ctx_p : (tag == 1) ? ctx_c : ctx_h;
  const float* Wc  = (tag == 0) ? Wc_p  : (tag == 1) ? Wc_c  : Wc_h;
  const float* bs  = (tag == 0) ? bs_p  : (tag == 1) ? bs_c  : bs_h;
  const float* bc  = (tag == 0) ? bc_p  : (tag == 1) ? bc_c  : bc_h;
  float acc = bs[a] + bc[a];
  for (int e = 0; e < E_DIM; ++e)
    acc = fmaf(ctx[e], Wc[e * A_DIM + a], acc);
  cbias[tag * A_DIM + a] = acc;
}

// ---------------------------------------------------------------------------
// K2: fused  e[tag][s] = sum_a tanh( (sentence @ W_sent)[s][a] + cbias[a] ) * v[a]
// One block = 32 rows x one tag. Sentence tile staged in LDS via TDM.
// ---------------------------------------------------------------------------
__global__ void attn_gemm_e_kernel(const float* __restrict__ sent,
                                   const float* __restrict__ Ws_p,
                                   const float* __restrict__ Ws_c,
                                   const float* __restrict__ Ws_h,
                                   const float* __restrict__ cbias,
                                   const float* __restrict__ v_p,
                                   const float* __restrict__ v_c,
                                   const float* __restrict__ v_h,
                                   float* __restrict__ e_buf) {
  extern __shared__ float smem[];
  float* e_sh = smem + TILE_DW;

  const int tag = blockIdx.x % 3;
  const int tile = blockIdx.x / 3;
  const int r0 = tile * M_TILE;
  const float* W = (tag == 0) ? Ws_p : (tag == 1) ? Ws_c : Ws_h;
  const float* v = (tag == 0) ? v_p : (tag == 1) ? v_c : v_h;

  const int tid = threadIdx.x;
  if (tid < M_TILE) e_sh[tid] = 0.0f;

#if defined(ATTN_HAVE_TDM)
  if (tid < 32) {  // wave 0 issues the tensor DMA (TDM ignores EXEC; per-wave op)
    unsigned long long ga =
        (unsigned long long)(const void*)(sent + (size_t)r0 * H2);
    u32x4 g0;
    g0.x = 1u;                                        // count=1, user mode, no gather
    g0.y = 0u;                                        // lds_addr = 0 (tile at LDS base)
    g0.z = (unsigned)(ga & 0xFFFFFFFFull);
    g0.w = (unsigned)((ga >> 32) & 0x01FFFFFFull) | 0x80000000u;  // type=2 ("image")
    i32x8 g1;
    const unsigned rows = (unsigned)(S_LEN - r0);
    g1[0] = (int)((2u << 16) | (1u << 20) | (7u << 22));  // data_size=4B, pad_en, ivl=256DW, amt=1DW
    g1[1] = (int)(((unsigned)H2 & 0xFFFFu) << 16);        // tensor_dim0 lo
    g1[2] = (int)(((unsigned)H2 >> 16) | ((rows & 0xFFFFu) << 16));  // dim0 hi | dim1 lo
    g1[3] = (int)((rows >> 16) | ((unsigned)H2 << 16));   // dim1 hi | tile_dim0=1024
    g1[4] = (int)M_TILE;                                  // tile_dim1=32, tile_dim2=0
    g1[5] = (int)H2;                                      // tensor_dim0_stride lo
    g1[6] = 0;
    g1[7] = 0;
    i32x4 gz4 = {0, 0, 0, 0};
    i32x8 gz8 = {0, 0, 0, 0, 0, 0, 0, 0};
    __builtin_amdgcn_tensor_load_to_lds(g0, g1, gz4, gz4, gz8, 0);
    __builtin_amdgcn_s_wait_tensorcnt(0);
  }
#else
  for (int L = tid; L < M_TILE * H2; L += 256)
    smem[L + (L >> 8)] = sent[(size_t)r0 * H2 + L];
#endif
  __syncthreads();

  const int lane = tid & 31;
  const int wv = tid >> 5;          // 8 waves
  const int half = lane >> 4;       // hi/lo 16 lanes
  const int l16 = lane & 15;
  const unsigned base0 = (unsigned)(l16 * LDS_ROW);             // A rows 0..15
  const unsigned base1 = (unsigned)((16 + l16) * LDS_ROW);      // A rows 16..31

  float e0[8] = {0.f, 0.f, 0.f, 0.f, 0.f, 0.f, 0.f, 0.f};
  float e1[8] = {0.f, 0.f, 0.f, 0.f, 0.f, 0.f, 0.f, 0.f};

  for (int nb = wv; nb < A_DIM / 16; nb += 8) {   // 16 column-chunks per wave
    const int n = nb * 16 + l16;
    const float* Wcol = W + n;
    v8f D0 = {};
    v8f D1 = {};
#pragma unroll 4
    for (int k = 0; k < H2; k += 4) {
      const int kb = k + half * 2;
      const int koff = kb + (kb >> 8);            // pad-aware dword offset in row
      v2f b;
      b.x = Wcol[kb * A_DIM];
      b.y = Wcol[(kb + 1) * A_DIM];
      v2f a0, a1;
      a0.x = smem[base0 + koff];
      a0.y = smem[base0 + koff + 1];
      a1.x = smem[base1 + koff];
      a1.y = smem[base1 + koff + 1];
      D0 = __builtin_amdgcn_wmma_f32_16x16x4_f32(false, a0, false, b, (short)0,
                                                 D0, false, false);
      D1 = __builtin_amdgcn_wmma_f32_16x16x4_f32(false, a1, false, b, (short)0,
                                                 D1, false, false);
    }
    const float cb = cbias[tag * A_DIM + n];      // col of this lane = lane%16
    const float vn = v[n];
#pragma unroll
    for (int r = 0; r < 8; ++r) {
      float t0 = ATTN_TANH(D0[r] + cb) * vn;
      float t1 = ATTN_TANH(D1[r] + cb) * vn;
      // reduce across the 16 column-lanes (stays inside each 16-lane group)
      t0 += __shfl_xor(t0, 1, 32); t0 += __shfl_xor(t0, 2, 32);
      t0 += __shfl_xor(t0, 4, 32); t0 += __shfl_xor(t0, 8, 32);
      t1 += __shfl_xor(t1, 1, 32); t1 += __shfl_xor(t1, 2, 32);
      t1 += __shfl_xor(t1, 4, 32); t1 += __shfl_xor(t1, 8, 32);
      e0[r] += t0;
      e1[r] += t1;
    }
  }

  if (l16 == 0) {  // one lane per 16-group holds each row's partial sum
#pragma unroll
    for (int r = 0; r < 8; ++r) {
      atomicAdd(&e_sh[half * 8 + r], e0[r]);        // rows r + 8*half
      atomicAdd(&e_sh[16 + half * 8 + r], e1[r]);   // rows 16 + r + 8*half
    }
  }
  __syncthreads();
  if (tid < M_TILE) e_buf[tag * S_LEN + r0 + tid] = e_sh[tid];
}

// ---------------------------------------------------------------------------
// K3: masked softmax per tag over S, fused_w = mean of the three weight vectors
// ---------------------------------------------------------------------------
__global__ void attn_softmax_kernel(const float* __restrict__ e_buf,
                                    const int* __restrict__ length,
                                    float* __restrict__ fw) {
  __shared__ float red[1024];
  const int tid = threadIdx.x;
  const int L = length[0];
  float acc[8] = {0.f, 0.f, 0.f, 0.f, 0.f, 0.f, 0.f, 0.f};

  for (int tag = 0; tag < 3; ++tag) {
    if (L > 0) {   // uniform branch; L<=0 => fully masked => weights are zero
      float ev[8];
      float lm = -INFINITY;
#pragma unroll
      for (int i = 0; i < 8; ++i) {
        const int s = tid + i * 1024;
        const float e = e_buf[tag * S_LEN + s];
        ev[i] = (s < L) ? e : -INFINITY;
        lm = fmaxf(lm, ev[i]);
      }
      red[tid] = lm;
      __syncthreads();
      for (int o = 512; o > 0; o >>= 1) {
        if (tid < o) red[tid] = fmaxf(red[tid], red[tid + o]);
        __syncthreads();
      }
      const float m = red[0];
      __syncthreads();
      float ls = 0.f;
#pragma unroll
      for (int i = 0; i < 8; ++i) {
        ev[i] = expf(ev[i] - m);   // masked lanes: exp(-inf) = 0
        ls += ev[i];
      }
      red[tid] = ls;
      __syncthreads();
      for (int o = 512; o > 0; o >>= 1) {
        if (tid < o) red[tid] += red[tid + o];
        __syncthreads();
      }
      const float inv = 1.0f / red[0];
      __syncthreads();
#pragma unroll
      for (int i = 0; i < 8; ++i) acc[i] += ev[i] * inv;
    }
  }
#pragma unroll
  for (int i = 0; i < 8; ++i)
    fw[tid + i * 1024] = acc[i] * (1.0f / 3.0f);
}

// ---------------------------------------------------------------------------
// K4: partial[sb][h] = sum over 256-row strip of fused_w[s]*sentence[s][h]
// K5: out[h] = sum_sb partial[sb][h]   (deterministic, no fp atomics)
// ---------------------------------------------------------------------------
__global__ void attn_wsum_kernel(const float* __restrict__ sent,
                                 const float* __restrict__ fw,
                                 float* __restrict__ partial) {
  const int h = ((blockIdx.x & 3) << 8) + threadIdx.x;
  const int s0 = (blockIdx.x >> 2) << 8;
  float acc = 0.f;
  for (int i = 0; i < 256; ++i)
    acc = fmaf(fw[s0 + i], sent[(size_t)(s0 + i) * H2 + h], acc);
  partial[(blockIdx.x >> 2) * H2 + h] = acc;
}

__global__ void attn_reduce_kernel(const float* __restrict__ partial,
                                   float* __restrict__ out) {
  const int h = (blockIdx.x << 8) + threadIdx.x;
  float acc = 0.f;
#pragma unroll
  for (int b = 0; b < 32; ++b) acc += partial[b * H2 + h];
  out[h] = acc;
}

// ---------------------------------------------------------------------------
extern "C" void kernel_launch(void* const* d_in, const int* in_sizes, int n_in,
                              void* d_out, int out_size, void* d_ws, size_t ws_size,
                              hipStream_t stream) {
  (void)in_sizes; (void)n_in; (void)out_size; (void)ws_size;
  const float* sent = (const float*)d_in[0];
  const int* length = (const int*)d_in[1];
  const float* ctx_p = (const float*)d_in[2];
  const float* ctx_c = (const float*)d_in[3];
  const float* ctx_h = (const float*)d_in[4];
  // per tag: W_sent, b_sent, W_ctx, b_ctx, v, bv  at base 5 + 6*tag
  const float* Ws_p = (const float*)d_in[5];
  const float* bs_p = (const float*)d_in[6];
  const float* Wc_p = (const float*)d_in[7];
  const float* bc_p = (const float*)d_in[8];
  const float* v_p  = (const float*)d_in[9];
  const float* Ws_c = (const float*)d_in[11];
  const float* bs_c = (const float*)d_in[12];
  const float* Wc_c = (const float*)d_in[13];
  const float* bc_c = (const float*)d_in[14];
  const float* v_c  = (const float*)d_in[15];
  const float* Ws_h = (const float*)d_in[17];
  const float* bs_h = (const float*)d_in[18];
  const float* Wc_h = (const float*)d_in[19];
  const float* bc_h = (const float*)d_in[20];
  const float* v_h  = (const float*)d_in[21];
  // (bv_* at 10/16/22 are unused: softmax is shift-invariant and fully-masked rows are zeroed)

  float* cbias   = (float*)d_ws;                 // 3*2048
  float* e_buf   = cbias + 3 * A_DIM;            // 3*8192
  float* fw      = e_buf + 3 * S_LEN;            // 8192
  float* partial = fw + S_LEN;                   // 32*1024
  float* out = (float*)d_out;

  (void)hipFuncSetAttribute((const void*)attn_gemm_e_kernel,
                            hipFuncAttributeMaxDynamicSharedMemorySize,
                            K2_LDS_BYTES);

  attn_cbias_kernel<<<dim3(24), dim3(256), 0, stream>>>(
      ctx_p, ctx_c, ctx_h, Wc_p, Wc_c, Wc_h,
      bs_p, bs_c, bs_h, bc_p, bc_c, bc_h, cbias);

  attn_gemm_e_kernel<<<dim3((S_LEN / M_TILE) * 3), dim3(256), K2_LDS_BYTES,
                       stream>>>(sent, Ws_p, Ws_c, Ws_h, cbias, v_p, v_c, v_h,
                                 e_buf);

  attn_softmax_kernel<<<dim3(1), dim3(1024), 0, stream>>>(e_buf, length, fw);

  attn_wsum_kernel<<<dim3((S_LEN / 256) * (H2 / 256)), dim3(256), 0, stream>>>(
      sent, fw, partial);

  attn_reduce_kernel<<<dim3(H2 / 256), dim3(256), 0, stream>>>(partial, out);
}